// GAESolution_88880053223616
// MI455X (gfx1250) — compile-verified
//
#include <hip/hip_runtime.h>
#include <stdint.h>

// ============================================================================
// GAE on MI455X (gfx1250), B=4096 x S=4096 f32.
//
// Memory-bound: 256 MiB mandatory traffic -> ~11us @ 23.3 TB/s. The backward
// recurrence A_t = delta_t + c*A_{t+1} (c = gamma*lam) is parallelized as a
// blocked linear scan:
//   * 16-step local suffix scans  == delta-tile (16b x 16t) x triangular
//     c-power matrix  -> V_WMMA_F32_16X16X4_F32 chained 4x over K.
//   * per-chunk carries combined serially with factor c^16 (32 steps/segment)
//   * elementwise fixup A = localA + c^(16-n) * carry, returns = A + values.
// One workgroup (256 thr = 8 waves) owns 16 batch rows; S processed in 8
// segments of 512 so all LDS state (~52 KB) stays comfortably resident.
// Staging uses gfx1250 async global->LDS copies (ASYNCcnt path).
// ============================================================================

#define G_GAMMA 0.99f
#define G_LAM   0.95f

typedef __attribute__((ext_vector_type(2))) float v2f;
typedef __attribute__((ext_vector_type(8))) float v8f;

#define ROWS   16              // batch rows per workgroup
#define SEG    512             // timesteps per resident segment
#define TILE   128             // staging tile width
#define CHUNK  16              // WMMA scan chunk (one 16x16 tile)
#define VSTR   132             // v_s/delta_s row stride (pad: bank step 4/row)
#define LASTR  516             // localA row stride (SEG+4 pad)
#define NCHUNK (SEG / CHUNK)   // 32
#define NTILE  (SEG / TILE)    // 4
#define TPB    256

#if defined(__HIP_DEVICE_COMPILE__) && __has_builtin(__builtin_amdgcn_wmma_f32_16x16x4_f32)
#define USE_WMMA 1
#endif
#if defined(__HIP_DEVICE_COMPILE__) && \
    __has_builtin(__builtin_amdgcn_global_load_async_to_lds_b32) && \
    __has_builtin(__builtin_amdgcn_s_wait_asynccnt)
#define USE_ASYNC 1
#endif

#if USE_ASYNC
typedef __attribute__((address_space(1))) int g1int;
typedef __attribute__((address_space(3))) int l3int;
#endif

__global__ __launch_bounds__(TPB)
void gae_kernel(const float* __restrict__ rewards,
                const float* __restrict__ values,
                const float* __restrict__ next_values,
                float* __restrict__ adv_out,
                float* __restrict__ ret_out,
                int B, int S)
{
    __shared__ float v_s    [ROWS * VSTR];     // staged values tile (TILE+1 cols)
    __shared__ float delta_s[ROWS * VSTR];     // delta tile
    __shared__ float localA [ROWS * LASTR];    // per-segment local suffix scans
    __shared__ float carryNx[ROWS * NCHUNK];   // A at start of next chunk
    __shared__ float carryA [ROWS];            // A at start of previously done segment
    __shared__ float powc_s [20];              // c^0 .. c^16

    const int tid     = threadIdx.x;
    const int lane    = tid & 31;              // wave32
    const int wave    = tid >> 5;              // 8 waves
    const int rowBase = blockIdx.x * ROWS;
    const float c     = G_GAMMA * G_LAM;

    if (tid < 17) { float p = 1.0f; for (int i = 0; i < tid; ++i) p *= c; powc_s[tid] = p; }
    if (tid < ROWS) carryA[tid] = 0.0f;
    __syncthreads();
    const float c16 = powc_s[16];

#if USE_WMMA
    // Constant B operands: T[k][n] = c^(k-n) for k>=n else 0 (4x16 slice per kb).
    // B layout (4x16, 2 VGPRs, wave32): lanes 0-15 -> N=lane, K = 4kb+{0,1};
    // lanes 16-31 -> N=lane-16, K = 4kb+{2,3}  (mirrors documented A/B striping).
    const int bn  = lane & 15;
    const int bhi = lane >> 4;
    v2f bop[4];
#pragma unroll
    for (int kb = 0; kb < 4; ++kb) {
        int k0 = kb * 4 + bhi * 2;
        bop[kb].x = (k0     >= bn) ? powc_s[k0     - bn] : 0.0f;
        bop[kb].y = (k0 + 1 >= bn) ? powc_s[k0 + 1 - bn] : 0.0f;
    }
#endif

    const int nseg = S / SEG;
    for (int seg = nseg - 1; seg >= 0; --seg) {
        const int c0seg = seg * SEG;

        // ---------------- Phase 1: local chunk scans for this segment --------
        for (int tt = 0; tt < NTILE; ++tt) {
            const int c0 = c0seg + tt * TILE;
            __syncthreads();   // v_s/delta_s free; prev segment's localA consumed

            // Stage values[b, c0 .. c0+TILE-1] -> LDS (coalesced along t).
            for (int idx = tid; idx < ROWS * TILE; idx += TPB) {
                int b = idx >> 7, col = idx & (TILE - 1);
                size_t g = (size_t)(rowBase + b) * S + c0 + col;
                __builtin_prefetch(rewards + g, 0, 1);     // warm the delta pass
#if USE_ASYNC
                __builtin_amdgcn_global_load_async_to_lds_b32(
                    (g1int*)(values + g),
                    (l3int*)&v_s[b * VSTR + col],
                    0, 0);
#else
                v_s[b * VSTR + col] = values[g];
#endif
            }
#if USE_ASYNC
            __builtin_amdgcn_s_wait_asynccnt(0);
#endif
            // +1 boundary column: values[., c0+TILE] or next_values bootstrap.
            if (tid < ROWS) {
                int t = c0 + TILE;
                v_s[tid * VSTR + TILE] =
                    (t < S) ? values[(size_t)(rowBase + tid) * S + t]
                            : next_values[rowBase + tid];
            }
            __syncthreads();

            // delta = r + gamma*v[t+1] - v[t]  (rewards streamed coalesced)
            for (int idx = tid; idx < ROWS * TILE; idx += TPB) {
                int b = idx >> 7, col = idx & (TILE - 1);
                float r = rewards[(size_t)(rowBase + b) * S + c0 + col];
                delta_s[b * VSTR + col] =
                    r + G_GAMMA * v_s[b * VSTR + col + 1] - v_s[b * VSTR + col];
            }
            __syncthreads();

            // Each wave computes one 16-wide chunk: localA = delta_tile x T.
            {
                const int dcol = wave * CHUNK;             // col within tile
                const int ccol = tt * TILE + dcol;         // col within segment
#if USE_WMMA
                const int row = lane & 15;                 // A-matrix M = lane%16
                const int hi  = lane >> 4;
                v8f acc = {};
#pragma unroll
                for (int kb = 0; kb < 4; ++kb) {
                    int k0 = kb * 4 + hi * 2;              // A: K=4kb+{0,1}/{2,3}
                    v2f a = *(const v2f*)&delta_s[row * VSTR + dcol + k0];
                    acc = __builtin_amdgcn_wmma_f32_16x16x4_f32(
                        false, a, false, bop[kb], (short)0, acc, false, false);
                }
                // D layout: VGPR j -> M=j (lanes 0-15) / M=j+8 (lanes 16-31), N=lane%16
#pragma unroll
                for (int j = 0; j < 8; ++j)
                    localA[(j + 8 * hi) * LASTR + ccol + (lane & 15)] = acc[j];
#else
                // Scalar fallback: O(16) suffix sum per element.
                for (int e = lane; e < ROWS * CHUNK; e += 32) {
                    int b = e >> 4, nn = e & 15;
                    float a = 0.0f;
                    for (int k = 15; k >= nn; --k)
                        a = delta_s[b * VSTR + dcol + k] + c * a;
                    localA[b * LASTR + ccol + nn] = a;
                }
#endif
            }
        }
        __syncthreads();   // all localA for the segment ready

        // ---------------- Phase 2: serial chunk-carry combine ----------------
        if (tid < ROWS) {
            float carry = carryA[tid];                     // A at start of next segment
            for (int i = NCHUNK - 1; i >= 0; --i) {
                carryNx[tid * NCHUNK + i] = carry;         // A at start of chunk i+1
                carry = localA[tid * LASTR + i * CHUNK] + c16 * carry;
            }
            carryA[tid] = carry;                           // A at start of this segment
        }
        __syncthreads();

        // ---------------- Phase 3: fixup + coalesced stores -------------------
        for (int idx = tid; idx < ROWS * SEG; idx += TPB) {
            int b  = idx >> 9;                             // SEG == 512
            int tl = idx & (SEG - 1);
            float carry = carryNx[b * NCHUNK + (tl >> 4)];
            float A = localA[b * LASTR + tl] + powc_s[16 - (tl & 15)] * carry;
            size_t g = (size_t)(rowBase + b) * S + c0seg + tl;
            float v = values[g];                           // L2-resident re-read
            adv_out[g] = A;
            ret_out[g] = A + v;
        }
        // next segment's first __syncthreads() protects localA/carryNx reuse
    }
}

extern "C" void kernel_launch(void* const* d_in, const int* in_sizes, int n_in,
                              void* d_out, int out_size, void* d_ws, size_t ws_size,
                              hipStream_t stream) {
    (void)n_in; (void)out_size; (void)d_ws; (void)ws_size;
    const float* rewards     = (const float*)d_in[0];
    const float* values      = (const float*)d_in[1];
    const float* next_values = (const float*)d_in[2];
    const int B = in_sizes[2];                 // next_values length
    const int S = in_sizes[0] / B;             // 4096; assumed multiple of SEG
    float* adv = (float*)d_out;
    float* ret = adv + (size_t)B * S;
    dim3 grid((unsigned)(B / ROWS));           // 256 workgroups
    gae_kernel<<<grid, TPB, 0, stream>>>(rewards, values, next_values, adv, ret, B, S);
}